// GMNet_1726576854585
// MI455X (gfx1250) — compile-verified
//
#include <hip/hip_runtime.h>
#include <math.h>

// ---------------------------------------------------------------------------
// CDNA5 / gfx1250 implementation of GMNet forward:
//   enc = relu(relu(X @ W1^T + b1) @ W2^T + b2)        (f32 WMMA GEMMs)
//   M   = e1 @ e2^T ; V = softmax(200*M) ; p = -V      (f32 WMMA + LDS)
//   z   = structured PDIPM (Sherman-Morrison + 16x16 Schur solve)
// Precision: fp32 end-to-end via V_WMMA_F32_16X16X4_F32 (problem is
// latency-bound, not throughput-bound on MI455X; softmax(200*x) + IPM punish
// low precision, so the f32 matrix pipe is the right choice here).
// ---------------------------------------------------------------------------

typedef __attribute__((ext_vector_type(2))) float v2f;
typedef __attribute__((ext_vector_type(8))) float v8f;

// ---------------------------------------------------------------------------
// GEMM: Y[m,n] = relu( sum_k X[m,k]*W[n,k] + bias[n] )
// X: (Mrows x K) row-major, W: (512 x K) row-major, Y: (Mrows x 512)
// One 16x16 output tile per wave, 8 waves per block.
// A fragment (16x4 f32, ISA 7.12.2):  lane(m=l&15,g=l>>4) -> A[m][k0+2g..+1]
// B fragment (4x16  f32, mirrored):   lane(n=l&15,g=l>>4) -> W[n][k0+2g..+1]
// D (16x16 f32): lane(n=l&15), VGPR r -> row (l>>4)*8 + r
// ---------------------------------------------------------------------------
__global__ __launch_bounds__(256)
void gemm_relu_wmma_f32(const float* __restrict__ X, const float* __restrict__ W,
                        const float* __restrict__ bias, float* __restrict__ Y,
                        int K)
{
    const int lane = threadIdx.x & 31;
    const int wave = threadIdx.x >> 5;
    const int row0 = blockIdx.x * 16;
    const int col0 = (blockIdx.y * 8 + wave) * 16;

    const int m = lane & 15;      // A row (== B "n" for our load pattern)
    const int g = lane >> 4;      // K sub-group

    const float* arow = X + (size_t)(row0 + m) * (size_t)K + 2 * g;
    const float* brow = W + (size_t)(col0 + m) * (size_t)K + 2 * g;

    v8f c = {};
#pragma unroll 8
    for (int k0 = 0; k0 < K; k0 += 4) {
        v2f a = *(const v2f*)(arow + k0);
        v2f b = *(const v2f*)(brow + k0);
        c = __builtin_amdgcn_wmma_f32_16x16x4_f32(
                false, a, false, b, (short)0, c, false, false);
    }

    const int n  = lane & 15;
    const float bv = bias[col0 + n];
    const int mb = g * 8;
#pragma unroll
    for (int r = 0; r < 8; ++r) {
        float v = c[r] + bv;
        v = v > 0.0f ? v : 0.0f;
        Y[(size_t)(row0 + mb + r) * 512 + col0 + n] = v;
    }
}

// ---------------------------------------------------------------------------
// Fused per-batch kernel: M = E1_b @ E2_b^T (WMMA), softmax, structured PDIPM.
// One block (256 threads) per batch element; thread t <-> z entry (i,j).
//
// QP structure (n=16, nz=256): G = [-I; C], A = row sums.
//   K11 = I + diag(d1) + C^T diag(d2) C  is block-diagonal per column j:
//       B_j = diag(1+d1[:,j]) + d2_j * 1 1^T   (Sherman-Morrison inverse)
//   Schur: S = A K11^{-1} A^T is dense 16x16 SPD -> tiny Gaussian solve.
// ---------------------------------------------------------------------------
__global__ __launch_bounds__(256)
void match_pdipm(const float* __restrict__ E1, const float* __restrict__ E2,
                 float* __restrict__ out)
{
    __shared__ float sM[256];          // M and general scratch
    __shared__ float zsh[256];
    __shared__ float red[256];
    __shared__ float invd[256];
    __shared__ float Smat[16 * 17];    // [.,16] = rhs column
    __shared__ float s2[16], l2[16], nu_[16], d2[16], w2[16], ri2[16];
    __shared__ float req_[16], ds2[16], dl2[16], colz[16], rowz[16];
    __shared__ float coldz[16], cfac[16], t1[16], dnu_[16], rmax[16], rsum[16];

    const int t = threadIdx.x;
    const int b = blockIdx.x;
    const int i = t >> 4;
    const int j = t & 15;

    // ---- M = E1_b(16x512) @ E2_b(16x512)^T via wave 0 WMMA ----
    if (t < 32) {
        const int lane = t;
        const int m = lane & 15, g = lane >> 4;
        const float* arow = E1 + ((size_t)b * 16 + m) * 512 + 2 * g;
        const float* brow = E2 + ((size_t)b * 16 + m) * 512 + 2 * g;
        v8f c = {};
#pragma unroll 8
        for (int k0 = 0; k0 < 512; k0 += 4) {
            v2f a  = *(const v2f*)(arow + k0);
            v2f bb = *(const v2f*)(brow + k0);
            c = __builtin_amdgcn_wmma_f32_16x16x4_f32(
                    false, a, false, bb, (short)0, c, false, false);
        }
        const int n = lane & 15, mb = g * 8;
#pragma unroll
        for (int r = 0; r < 8; ++r) sM[(mb + r) * 16 + n] = c[r];
    }
    __syncthreads();

    // ---- p = -softmax(ALPHA * M) over columns j (rows of V) ----
    if (t < 16) {
        float mx = -1e30f;
        for (int q = 0; q < 16; ++q) mx = fmaxf(mx, sM[t * 16 + q]);
        rmax[t] = mx;
    }
    __syncthreads();
    const float e = __expf(200.0f * (sM[i * 16 + j] - rmax[i]));
    zsh[t] = e;
    __syncthreads();
    if (t < 16) {
        float sum = 0.0f;
        for (int q = 0; q < 16; ++q) sum += zsh[t * 16 + q];
        rsum[t] = sum;
    }
    __syncthreads();
    const float p = -e / rsum[i];

    // ---- PDIPM state: z = 1/32, s1 = z, lam = 1, s2 = 0.5, nu = 0 ----
    float z  = 1.0f / 32.0f;
    float s1 = 1.0f / 32.0f;
    float l1 = 1.0f;
    if (t < 16) { s2[t] = 0.5f; l2[t] = 1.0f; nu_[t] = 0.0f; }
    __syncthreads();

    for (int iter = 0; iter < 10; ++iter) {
        // mu = mean(s*lam) over 272 entries
        red[t] = s1 * l1 + ((t < 16) ? s2[t] * l2[t] : 0.0f);
        __syncthreads();
        for (int off = 128; off > 0; off >>= 1) {
            if (t < off) red[t] += red[t + off];
            __syncthreads();
        }
        const float mu = red[0] / 272.0f;
        __syncthreads();

        const float d1 = l1 / s1;
        if (t < 16) d2[t] = l2[t] / s2[t];

        // column / row sums of z
        zsh[t] = z;
        __syncthreads();
        if (t < 16) {
            float cs = 0.0f, rs = 0.0f;
            for (int q = 0; q < 16; ++q) {
                cs += zsh[q * 16 + t];
                rs += zsh[t * 16 + q];
            }
            colz[t] = cs; rowz[t] = rs;
            req_[t] = rs - 1.0f;
            ri2[t]  = cs + s2[t] - 1.0f;
        }
        __syncthreads();

        const float rz  = z + p - l1 + l2[j] + nu_[i];   // TAU = 1
        const float ri1 = -z + s1;
        const float rc1 = l1 * s1 - 0.1f * mu;           // SIGMA = 0.1
        const float w1  = d1 * ri1 - rc1 / s1;
        if (t < 16) {
            const float rc2 = l2[t] * s2[t] - 0.1f * mu;
            w2[t] = d2[t] * ri2[t] - rc2 / s2[t];
        }
        __syncthreads();
        const float rhs = -(rz - w1 + w2[j]);

        // K11 column-block diagonals and Sherman-Morrison factors
        const float iv = 1.0f / (1.0f + d1);
        invd[t] = iv;
        __syncthreads();
        if (t < 16) {
            float sg = 0.0f;
            for (int q = 0; q < 16; ++q) sg += invd[q * 16 + t];
            cfac[t] = d2[t] / (1.0f + d2[t] * sg);
        }
        __syncthreads();

        // S = A K11^{-1} A^T : thread t handles (i, i2=j)
        {
            const int i2 = j;
            float acc = 0.0f;
            for (int q = 0; q < 16; ++q) {
                const float a1 = invd[i * 16 + q];
                const float a2 = invd[i2 * 16 + q];
                acc -= cfac[q] * a1 * a2;
                if (i == i2) acc += a1;
            }
            Smat[i * 17 + i2] = acc;
        }

        // rK = A K11^{-1} rhs + req  -> Smat[:,16]
        red[t] = iv * rhs;
        __syncthreads();
        if (t < 16) {
            float tt = 0.0f;
            for (int q = 0; q < 16; ++q) tt += red[q * 16 + t];
            t1[t] = tt;
        }
        __syncthreads();
        red[t] = iv * (rhs - cfac[j] * t1[j]);   // u = K11^{-1} rhs
        __syncthreads();
        if (t < 16) {
            float rsu = 0.0f;
            for (int q = 0; q < 16; ++q) rsu += red[t * 16 + q];
            Smat[t * 17 + 16] = rsu + req_[t];
        }
        __syncthreads();

        // 16x16 SPD solve: forward elimination (no pivot) + back-sub
        for (int k = 0; k < 16; ++k) {
            if (t < 16 && t > k) {
                const float f = Smat[t * 17 + k] / Smat[k * 17 + k];
                for (int cI = k; cI <= 16; ++cI)
                    Smat[t * 17 + cI] -= f * Smat[k * 17 + cI];
            }
            __syncthreads();
        }
        if (t == 0) {
            for (int k = 15; k >= 0; --k) {
                float acc = Smat[k * 17 + 16];
                for (int cI = k + 1; cI < 16; ++cI)
                    acc -= Smat[k * 17 + cI] * dnu_[cI];
                dnu_[k] = acc / Smat[k * 17 + k];
            }
        }
        __syncthreads();

        // dz = K11^{-1} (rhs - A^T dnu)
        const float v = rhs - dnu_[i];
        red[t] = iv * v;
        __syncthreads();
        if (t < 16) {
            float tt = 0.0f;
            for (int q = 0; q < 16; ++q) tt += red[q * 16 + t];
            t1[t] = tt;
        }
        __syncthreads();
        const float dz = iv * (v - cfac[j] * t1[j]);

        // column sums of dz
        zsh[t] = dz;
        __syncthreads();
        if (t < 16) {
            float cs = 0.0f;
            for (int q = 0; q < 16; ++q) cs += zsh[q * 16 + t];
            coldz[t] = cs;
        }
        __syncthreads();

        const float ds1 = -ri1 + dz;
        const float dl1 = (-rc1 - l1 * ds1) / s1;
        if (t < 16) {
            ds2[t] = -ri2[t] - coldz[t];
            const float rc2 = l2[t] * s2[t] - 0.1f * mu;
            dl2[t] = (-rc2 - l2[t] * ds2[t]) / s2[t];
        }
        __syncthreads();

        // alpha = min(1, 0.99 * min over all fraction-to-boundary ratios)
        float rmv = fminf((ds1 < 0.0f) ? (-s1 / ds1) : 1e9f,
                          (dl1 < 0.0f) ? (-l1 / dl1) : 1e9f);
        if (t < 16) {
            rmv = fminf(rmv, (ds2[t] < 0.0f) ? (-s2[t] / ds2[t]) : 1e9f);
            rmv = fminf(rmv, (dl2[t] < 0.0f) ? (-l2[t] / dl2[t]) : 1e9f);
        }
        red[t] = rmv;
        __syncthreads();
        for (int off = 128; off > 0; off >>= 1) {
            if (t < off) red[t] = fminf(red[t], red[t + off]);
            __syncthreads();
        }
        const float alpha = fminf(1.0f, 0.99f * red[0]);
        __syncthreads();

        z  += alpha * dz;
        s1 += alpha * ds1;
        l1 += alpha * dl1;
        if (t < 16) {
            s2[t] += alpha * ds2[t];
            l2[t] += alpha * dl2[t];
            nu_[t] += alpha * dnu_[t];
        }
        __syncthreads();
    }

    out[(size_t)b * 256 + t] = z;
}

// ---------------------------------------------------------------------------
extern "C" void kernel_launch(void* const* d_in, const int* in_sizes, int n_in,
                              void* d_out, int out_size, void* d_ws, size_t ws_size,
                              hipStream_t stream)
{
    (void)in_sizes; (void)n_in; (void)out_size; (void)ws_size;

    const float* feat1 = (const float*)d_in[0];   // (64,16,512)
    const float* feat2 = (const float*)d_in[1];   // (64,16,512)
    const float* W1    = (const float*)d_in[2];   // (512,512)
    const float* b1    = (const float*)d_in[3];   // (512)
    const float* W2    = (const float*)d_in[4];   // (512,512)
    const float* b2    = (const float*)d_in[5];   // (512)
    float* out = (float*)d_out;                   // (64,16,16) fp32

    float* ws = (float*)d_ws;
    const size_t RK = 1024 * 512;                 // elements per intermediate
    float* H1 = ws;
    float* H2 = ws + RK;
    float* E1 = ws + 2 * RK;
    float* E2 = ws + 3 * RK;

    dim3 blk(256);
    dim3 grd(64, 4);  // 64 row-tiles x (4*8 waves) = 64x32 tiles of 16x16

    gemm_relu_wmma_f32<<<grd, blk, 0, stream>>>(feat1, W1, b1, H1, 512);
    gemm_relu_wmma_f32<<<grd, blk, 0, stream>>>(feat2, W1, b1, H2, 512);
    gemm_relu_wmma_f32<<<grd, blk, 0, stream>>>(H1,    W2, b2, E1, 512);
    gemm_relu_wmma_f32<<<grd, blk, 0, stream>>>(H2,    W2, b2, E2, 512);

    match_pdipm<<<dim3(64), blk, 0, stream>>>(E1, E2, out);
}